// AntiAliasActivation_6021544149506
// MI455X (gfx1250) — compile-verified
//
#include <hip/hip_runtime.h>

// AntiAliasActivation fused kernel for MI455X (gfx1250, wave32).
// x:(16,512,4096) f32 -> 2x upsample (12-tap) -> snake -> 2x downsample (12-tap) -> (16,512,4096) f32
// Memory-bound: >=256MiB HBM traffic -> ~11.5us floor @ 23.3 TB/s. All intermediates live in LDS.
// Downsample conv done as D(16x16) = W(16x44) x Z(44x16) via 11x v_wmma_f32_16x16x4_f32 per group.
// z stored in a bank-swizzled LDS layout (pad 4 dwords per 64) so the stride-32 B-fragment
// gathers are conflict-free instead of 8-way serialized.

typedef __attribute__((ext_vector_type(2))) float v2f;
typedef __attribute__((ext_vector_type(8))) float v8f;

#define T_LEN 4096
#define S_LEN 8192
#define NT    256

#define XOFF  4                       // left halo of xs (edge replicated)
#define XS_SZ (T_LEN + 8)             // 4 left + 4 right halo
#define ZOFF  8                       // left halo of zs (logical)
#define ZS_SZ 8720                    // logical 8207 + 4*(8207>>6) swizzle pad

// bank swizzle: insert 4 pad dwords every 64 logical dwords
__device__ __forceinline__ int zsw(int l) { return l + ((l >> 6) << 2); }

__global__ __launch_bounds__(NT) void aa_act_kernel(
    const float* __restrict__ x,
    const float* __restrict__ alpha,
    const float* __restrict__ beta,
    const float* __restrict__ uf,
    const float* __restrict__ df,
    float* __restrict__ out)
{
    __shared__ __attribute__((aligned(16))) float xs[XS_SZ];
    __shared__ __attribute__((aligned(16))) float zs[ZS_SZ];
    __shared__ float uf_s[12];
    __shared__ float df_s[12];

    const int tid = threadIdx.x;
    const int row = blockIdx.x;          // row = b*512 + c
    const int c   = row & 511;

    if (tid < 12) { uf_s[tid] = uf[tid]; df_s[tid] = df[tid]; }

    // per-channel snake params (uniform per block -> scalar path)
    const float ea = __expf(alpha[c]);
    const float cb = 1.0f / (__expf(beta[c]) + 1e-9f);

    // ---- phase 1: stage x row into LDS (+ edge halo), float4 coalesced ----
    const float* xrow = x + (size_t)row * T_LEN;
    const float4* xg  = (const float4*)xrow;
    float4*       xs4 = (float4*)(xs + XOFF);
    #pragma unroll
    for (int k = 0; k < 4; ++k) {
        int i = tid + k * NT;            // 0..1023 float4s
        xs4[i] = xg[i];
    }
    if (tid < 4) {                       // edge-pad halos
        xs[tid]                = xrow[0];
        xs[XOFF + T_LEN + tid] = xrow[T_LEN - 1];
    }
    __syncthreads();

    // ---- phase 2: 2x upsample (12-tap, edge pad) + snake -> zs (4 z per iter) ----
    // z[2m]   = snake( 2*sum_j uf[11-2j]*x_e[m-3+j] )
    // z[2m+1] = snake( 2*sum_j uf[10-2j]*x_e[m-2+j] )
    float ce[6], co[6];
    #pragma unroll
    for (int j = 0; j < 6; ++j) { ce[j] = uf_s[11 - 2*j]; co[j] = uf_s[10 - 2*j]; }

    #pragma unroll 2
    for (int k = 0; k < 8; ++k) {
        int M    = 2 * (tid + (k << 8));     // even, covers 0..4094; pair (M, M+1)
        int base = M - 4 + XOFF;             // == M, even -> b64-aligned
        float xv[10];                        // x_e[M-4 .. M+5]
        #pragma unroll
        for (int p = 0; p < 5; ++p) {        // 5x ds_load_b64, conflict-free
            float2 t = *(const float2*)(xs + base + 2 * p);
            xv[2*p] = t.x; xv[2*p + 1] = t.y;
        }
        float y0 = 0.f, y1 = 0.f, y2 = 0.f, y3 = 0.f;
        #pragma unroll
        for (int j = 0; j < 6; ++j) {
            y0 = __builtin_fmaf(ce[j], xv[j + 1], y0);   // m=M,   even tap
            y1 = __builtin_fmaf(co[j], xv[j + 2], y1);   // m=M,   odd tap
            y2 = __builtin_fmaf(ce[j], xv[j + 2], y2);   // m=M+1, even tap
            y3 = __builtin_fmaf(co[j], xv[j + 3], y3);   // m=M+1, odd tap
        }
        y0 *= 2.0f; y1 *= 2.0f; y2 *= 2.0f; y3 *= 2.0f;
        float s0 = __sinf(y0 * ea), s1 = __sinf(y1 * ea);
        float s2 = __sinf(y2 * ea), s3 = __sinf(y3 * ea);
        float z0 = __builtin_fmaf(cb * s0, s0, y0);
        float z1 = __builtin_fmaf(cb * s1, s1, y1);
        float z2 = __builtin_fmaf(cb * s2, s2, y2);
        float z3 = __builtin_fmaf(cb * s3, s3, y3);

        int L = 2 * M + ZOFF;                // == 0 mod 4 -> swizzled addr stays 16B-aligned
        float4 zq = { z0, z1, z2, z3 };
        *(float4*)(zs + zsw(L)) = zq;        // ds_store_b128, conflict-free

        if (tid == 0 && k == 0) {            // left edge halo: z[-1..] = z[0]
            #pragma unroll
            for (int i = 0; i < ZOFF; ++i) zs[i] = z0;   // zsw(i)==i for i<64
        }
        if (tid == (NT - 1) && k == 7) {     // right edge halo: z[8192..] = z[8191]
            #pragma unroll
            for (int i = 0; i < 7; ++i) zs[zsw(S_LEN + ZOFF + i)] = z3;
        }
    }

    // ---- constant A fragments: W[r,p] = df[p-2r], 0<=p-2r<12, K padded 42->44 ----
    // A 16x4 f32 layout (ISA 7.12.2): M = lane%16; VGPR0 holds K={0|2}, VGPR1 K={1|3} per lane half.
    const int lane = tid & 31;
    const int wave = tid >> 5;
    const int r    = lane & 15;
    const int koff = (lane < 16) ? 0 : 2;
    v2f aW[11];
    #pragma unroll
    for (int q = 0; q < 11; ++q) {
        int p0 = 4 * q + koff;
        int i0 = p0 - 2 * r;
        int i1 = i0 + 1;
        aW[q].x = (i0 >= 0 && i0 < 12) ? df_s[i0] : 0.0f;
        aW[q].y = (i1 >= 0 && i1 < 12) ? df_s[i1] : 0.0f;
    }
    __syncthreads();

    // ---- phase 3: stride-2 downsample as WMMA banded GEMM ----
    // out[16*tau + r] = sum_p W[r,p] * z_e[32*tau - 5 + p]; 16 taus (columns) per WMMA group.
    // B gather logical index = (32*tau - 5 + p) + ZOFF; swizzle kills the stride-32 bank conflicts.
    float* orow = out + (size_t)row * T_LEN;
    #pragma unroll
    for (int gg = 0; gg < 2; ++gg) {
        int g  = wave * 2 + gg;                  // 16 groups x 256 outputs = 4096
        int lb = 512 * g + 3 + 32 * r + koff;    // logical z index, q=0 (always in-bounds w/ halo)
        v8f acc = {};
        #pragma unroll
        for (int q = 0; q < 11; ++q) {
            int l0 = lb + 4 * q;
            v2f b;
            b.x = zs[zsw(l0)];
            b.y = zs[zsw(l0 + 1)];
            acc = __builtin_amdgcn_wmma_f32_16x16x4_f32(
                false, aW[q], false, b, (short)0, acc, false, false);
        }
        // D layout: VGPR v -> row M = v + (lane<16?0:8), col N = lane%16 (= tile index)
        int base = g * 256 + 16 * r + ((lane < 16) ? 0 : 8);
        float4 lo = { acc[0], acc[1], acc[2], acc[3] };
        float4 hi = { acc[4], acc[5], acc[6], acc[7] };
        *(float4*)(orow + base)     = lo;
        *(float4*)(orow + base + 4) = hi;
    }
}

extern "C" void kernel_launch(void* const* d_in, const int* in_sizes, int n_in,
                              void* d_out, int out_size, void* d_ws, size_t ws_size,
                              hipStream_t stream) {
    (void)in_sizes; (void)n_in; (void)out_size; (void)d_ws; (void)ws_size;
    const float* x     = (const float*)d_in[0];
    const float* alpha = (const float*)d_in[1];
    const float* beta  = (const float*)d_in[2];
    const float* uf    = (const float*)d_in[3];
    const float* df    = (const float*)d_in[4];
    float* out = (float*)d_out;

    dim3 grid(16 * 512);   // one block per (b,c) row; 6 blocks/WGP by LDS (51KB of 320KB)
    dim3 block(NT);        // 8 waves (wave32)
    hipLaunchKernelGGL(aa_act_kernel, grid, block, 0, stream,
                       x, alpha, beta, uf, df, out);
}